// GCN_89266600280762
// MI455X (gfx1250) — compile-verified
//
#include <hip/hip_runtime.h>
#include <hip/hip_bf16.h>

#define N_NODES   100000
#define N_EDGES   3200000
#define IN_DIM    512
#define HIDDEN    16
#define N_CLASSES 40
#define NCP       48   // N_CLASSES padded to 3 WMMA n-tiles

typedef __attribute__((ext_vector_type(2))) float v2f;
typedef __attribute__((ext_vector_type(8))) float v8f;

// ---------------------------------------------------------------------------
// Zero-fill (graph-capture-safe)
// ---------------------------------------------------------------------------
__global__ void gcn_zero_f32(float* __restrict__ p, int n) {
    int i = blockIdx.x * blockDim.x + threadIdx.x;
    int stride = gridDim.x * blockDim.x;
    for (; i < n; i += stride) p[i] = 0.0f;
}

// ---------------------------------------------------------------------------
// GEMM1: support1[100000,16] = x[100000,512] @ W1[512,16]  (fp32 WMMA)
// One wave per 16-row m-tile; W1 staged in LDS (32 KB); K-loop step 4.
// ---------------------------------------------------------------------------
__global__ __launch_bounds__(256) void gcn_gemm1_wmma(
    const float* __restrict__ x, const float* __restrict__ W1,
    float* __restrict__ support1) {
    __shared__ float w1s[IN_DIM * HIDDEN];
    for (int i = threadIdx.x; i < IN_DIM * HIDDEN; i += 256) w1s[i] = W1[i];
    __syncthreads();

    const int wave  = threadIdx.x >> 5;
    const int lane  = threadIdx.x & 31;
    const int mtile = blockIdx.x * 8 + wave;
    if (mtile >= N_NODES / 16) return;

    const int half = lane >> 4;   // 0: K=0,1  1: K=2,3
    const int l16  = lane & 15;
    const int row  = mtile * 16 + l16;
    const float* __restrict__ xr = x + (size_t)row * IN_DIM;

    v8f c = {};
    #pragma unroll 4
    for (int k = 0; k < IN_DIM; k += 4) {
        const int kk = k + 2 * half;
        v2f a;  // A 16x4 fragment: contiguous pair -> global_load_b64
        a.x = xr[kk];
        a.y = xr[kk + 1];
        v2f b;  // B 4x16 fragment from LDS
        b.x = w1s[kk * HIDDEN + l16];
        b.y = w1s[(kk + 1) * HIDDEN + l16];
        c = __builtin_amdgcn_wmma_f32_16x16x4_f32(
                false, a, false, b, (short)0, c, false, false);
    }

    // D layout: VGPR i, lanes 0-15 -> M=i, lanes 16-31 -> M=8+i
    float* __restrict__ out = support1 + (size_t)(mtile * 16 + 8 * half) * HIDDEN + l16;
    #pragma unroll
    for (int i = 0; i < 8; ++i) out[i * HIDDEN] = c[i];
}

// ---------------------------------------------------------------------------
// Scatter (16 cols): acc[dst] += src_feat[src] * w, optional fused ReLU on the
// gathered value (RELU=1 implements A_hat @ relu(h) for layer 2).
// One thread per edge; gathers are L2-resident (6.4 MB table in 192 MB L2).
// ---------------------------------------------------------------------------
template <int RELU>
__global__ __launch_bounds__(256) void gcn_scatter16(
    const int* __restrict__ es, const int* __restrict__ ed,
    const float* __restrict__ ew, const float* __restrict__ feat,
    float* __restrict__ acc) {
    const int e = blockIdx.x * 256 + threadIdx.x;
    if (e >= N_EDGES) return;
    const int s = es[e];
    const int d = ed[e];
    const float w = ew[e];
    const float4* __restrict__ sp = (const float4*)(feat + (size_t)s * HIDDEN);
    float* __restrict__ ap = acc + (size_t)d * HIDDEN;
    #pragma unroll
    for (int j = 0; j < 4; ++j) {
        float4 v = sp[j];
        if (RELU) {
            v.x = fmaxf(v.x, 0.0f); v.y = fmaxf(v.y, 0.0f);
            v.z = fmaxf(v.z, 0.0f); v.w = fmaxf(v.w, 0.0f);
        }
        atomicAdd(ap + 4 * j + 0, v.x * w);
        atomicAdd(ap + 4 * j + 1, v.y * w);
        atomicAdd(ap + 4 * j + 2, v.z * w);
        atomicAdd(ap + 4 * j + 3, v.w * w);
    }
}

// ---------------------------------------------------------------------------
// GEMM2: logits[100000,40] = agg2[100000,16] @ W2[16,40]  (fp32 WMMA)
// Reassociated: (A_hat @ relu(h)) @ W2 — plain stores, no atomics.
// N padded to 48 -> 3 n-tiles; W2 zero-padded in LDS.
// ---------------------------------------------------------------------------
__global__ __launch_bounds__(256) void gcn_gemm2_wmma(
    const float* __restrict__ agg2, const float* __restrict__ W2,
    float* __restrict__ logits) {
    __shared__ float w2s[HIDDEN * NCP];
    for (int i = threadIdx.x; i < HIDDEN * NCP; i += 256) {
        const int kk = i / NCP, col = i % NCP;
        w2s[i] = (col < N_CLASSES) ? W2[kk * N_CLASSES + col] : 0.0f;
    }
    __syncthreads();

    const int gw = blockIdx.x * 8 + (threadIdx.x >> 5);
    if (gw >= (N_NODES / 16) * 3) return;
    const int mtile = gw / 3;
    const int ntile = gw % 3;
    const int lane = threadIdx.x & 31;
    const int half = lane >> 4;
    const int l16  = lane & 15;
    const int row  = mtile * 16 + l16;
    const float* __restrict__ ar = agg2 + (size_t)row * HIDDEN;

    v8f c = {};
    #pragma unroll
    for (int k = 0; k < HIDDEN; k += 4) {
        const int kk = k + 2 * half;
        v2f a;
        a.x = ar[kk];
        a.y = ar[kk + 1];
        v2f b;
        b.x = w2s[kk * NCP + ntile * 16 + l16];
        b.y = w2s[(kk + 1) * NCP + ntile * 16 + l16];
        c = __builtin_amdgcn_wmma_f32_16x16x4_f32(
                false, a, false, b, (short)0, c, false, false);
    }

    const int col = ntile * 16 + l16;     // divergence only after WMMA
    if (col < N_CLASSES) {
        float* __restrict__ out =
            logits + (size_t)(mtile * 16 + 8 * half) * N_CLASSES + col;
        #pragma unroll
        for (int i = 0; i < 8; ++i) out[i * N_CLASSES] = c[i];
    }
}

// ---------------------------------------------------------------------------
// Softmax over 40 classes, one wave32 per row, in place on d_out.
// ---------------------------------------------------------------------------
__global__ __launch_bounds__(256) void gcn_softmax40(float* __restrict__ logits) {
    const int row = blockIdx.x * 8 + (threadIdx.x >> 5);
    if (row >= N_NODES) return;
    const int lane = threadIdx.x & 31;
    float* __restrict__ lp = logits + (size_t)row * N_CLASSES;

    const float NEG_INF = -__builtin_inff();
    float v0 = lp[lane];                                           // cols 0..31
    float v1 = (lane < N_CLASSES - 32) ? lp[32 + lane] : NEG_INF;  // cols 32..39

    float m = fmaxf(v0, v1);
    #pragma unroll
    for (int off = 16; off >= 1; off >>= 1)
        m = fmaxf(m, __shfl_xor(m, off, 32));

    float e0 = __expf(v0 - m);
    float e1 = (lane < N_CLASSES - 32) ? __expf(v1 - m) : 0.0f;
    float ssum = e0 + e1;
    #pragma unroll
    for (int off = 16; off >= 1; off >>= 1)
        ssum += __shfl_xor(ssum, off, 32);

    const float inv = 1.0f / ssum;
    lp[lane] = e0 * inv;
    if (lane < N_CLASSES - 32) lp[32 + lane] = e1 * inv;
}

// ---------------------------------------------------------------------------
extern "C" void kernel_launch(void* const* d_in, const int* in_sizes, int n_in,
                              void* d_out, int out_size, void* d_ws, size_t ws_size,
                              hipStream_t stream) {
    const float* x  = (const float*)d_in[0];
    const int*   es = (const int*)d_in[1];
    const int*   ed = (const int*)d_in[2];
    const float* ew = (const float*)d_in[3];
    const float* W1 = (const float*)d_in[4];
    const float* W2 = (const float*)d_in[5];
    float* out = (float*)d_out;   // logits written densely, softmax in place

    char* ws = (char*)d_ws;
    const size_t seg = (size_t)N_NODES * HIDDEN * 4;   // 6.4 MB
    float* support1 = (float*)(ws);            // x @ W1
    float* h_agg    = (float*)(ws + seg);      // A_hat @ support1 (pre-ReLU)
    float* agg2     = (float*)(ws + 2 * seg);  // A_hat @ relu(h_agg)

    // zero the two atomic accumulators
    gcn_zero_f32<<<512, 256, 0, stream>>>(h_agg, N_NODES * HIDDEN);
    gcn_zero_f32<<<512, 256, 0, stream>>>(agg2,  N_NODES * HIDDEN);

    // layer 1: dense GEMM then 16-col scatter-add
    gcn_gemm1_wmma<<<(N_NODES / 16 + 7) / 8, 256, 0, stream>>>(x, W1, support1);
    gcn_scatter16<0><<<(N_EDGES + 255) / 256, 256, 0, stream>>>(es, ed, ew, support1, h_agg);

    // layer 2 reassociated: scatter relu(h) (16 cols), then dense GEMM by W2
    gcn_scatter16<1><<<(N_EDGES + 255) / 256, 256, 0, stream>>>(es, ed, ew, h_agg, agg2);
    gcn_gemm2_wmma<<<((N_NODES / 16) * 3 + 7) / 8, 256, 0, stream>>>(agg2, W2, out);

    // softmax
    gcn_softmax40<<<(N_NODES + 7) / 8, 256, 0, stream>>>(out);
}